// AxisAttnPool1D_70746701300383
// MI455X (gfx1250) — compile-verified
//
#include <hip/hip_runtime.h>
#include <hip/hip_bf16.h>

typedef __attribute__((ext_vector_type(2))) float v2f;
typedef __attribute__((ext_vector_type(8))) float v8f;

#define EPS_RMS 1.1920929e-07f

// Problem shape (fixed by the reference): x is (B=4, D=256, H=64, W=2048) f32.
constexpr int Dd = 256;
constexpr int Hh = 64;
constexpr int Ww = 2048;
constexpr int RS = Hh * Ww;   // stride between consecutive d-rows (elements)

__global__ __launch_bounds__(256)
void axis_attn_pool_kernel(const float* __restrict__ x,
                           const float* __restrict__ norm_w,
                           const float* __restrict__ score_w,
                           float* __restrict__ out)
{
    __shared__ __align__(16) float cw_lds[Dd];   // norm_w * score_w
    __shared__ __align__(16) float nw_lds[Dd];   // norm_w
    __shared__ __align__(16) float s_lds[Ww];    // scores, then exp(s-max)
    __shared__ __align__(16) float rs_lds[Ww];   // rsqrt(ms+eps)
    __shared__ __align__(16) float g_lds[Ww];    // a[w]*rsqrt(ms+eps)
    __shared__ __align__(16) float red[256];

    const int tid = threadIdx.x;          // 0..255 (8 wave32)
    const int bh  = blockIdx.x;           // 0..255 = b*H + h
    const int b   = bh / Hh;
    const int h   = bh % Hh;
    const float* xb = x + (size_t)b * (Dd * RS) + (size_t)h * Ww;

    // Stage per-channel weights into LDS (one element per thread).
    cw_lds[tid] = norm_w[tid] * score_w[tid];
    nw_lds[tid] = norm_w[tid];
    __syncthreads();

    // ---------------- Phase 1: column sum-of-squares + weighted dot ----------
    // Threads own contiguous columns -> fully coalesced 128B/wave loads.
    for (int chunk = 0; chunk < Ww / 256; ++chunk) {
        const int w = chunk * 256 + tid;
        const float* col = xb + w;
        float ss = 0.0f, dt = 0.0f;
        #pragma unroll 4
        for (int d = 0; d < Dd; ++d) {
            float v = col[(size_t)d * RS];
            ss = fmaf(v, v, ss);
            dt = fmaf(v, cw_lds[d], dt);
        }
        float rs = rsqrtf(ss * (1.0f / (float)Dd) + EPS_RMS);
        s_lds[w]  = dt * rs;
        rs_lds[w] = rs;
    }
    __syncthreads();

    // ---------------- Phase 2: softmax over W (block-wide reductions) --------
    float lmax = -INFINITY;
    for (int i = tid; i < Ww; i += 256) lmax = fmaxf(lmax, s_lds[i]);
    red[tid] = lmax;
    __syncthreads();
    for (int sft = 128; sft > 0; sft >>= 1) {
        if (tid < sft) red[tid] = fmaxf(red[tid], red[tid + sft]);
        __syncthreads();
    }
    const float gmax = red[0];
    __syncthreads();

    float lsum = 0.0f;
    for (int i = tid; i < Ww; i += 256) {
        float e = expf(s_lds[i] - gmax);
        s_lds[i] = e;                 // stash exp
        lsum += e;
    }
    red[tid] = lsum;
    __syncthreads();
    for (int sft = 128; sft > 0; sft >>= 1) {
        if (tid < sft) red[tid] += red[tid + sft];
        __syncthreads();
    }
    const float inv_sum = 1.0f / red[0];
    __syncthreads();

    for (int i = tid; i < Ww; i += 256)
        g_lds[i] = s_lds[i] * inv_sum * rs_lds[i];
    __syncthreads();

    // ---------------- Phase 3: pooled[d] = nw[d] * sum_w g[w]*x[d,w] ----------
    // WMMA f32 16x16x4: A = 16 d-rows x 4 w (per ISA layout: lanes 0-15 hold
    // rows; VGPR slot = K 0/1, lane-half selects K 2/3). B = g[w0..w0+3]
    // broadcast into all 16 N columns -> every C column accumulates the same
    // mat-vec, so no cross-column reduction is needed at the end.
    const int wave  = tid >> 5;           // 0..7
    const int lane  = tid & 31;
    const int lrow  = lane & 15;          // matrix row within A / N within C
    const int khalf = lane >> 4;          // 0: K=0,1   1: K=2,3

    for (int gi = 0; gi < 2; ++gi) {
        const int d0 = (wave + gi * 8) * 16;                // 16 groups of 16 rows
        const float* arow = xb + (size_t)(d0 + lrow) * RS + 2 * khalf;
        v8f acc = {};
        for (int c = 0; c < Ww / 4; ++c) {
            v2f a  = *(const v2f*)(arow + c * 4);                    // global, L2-hot
            v2f bb = *(const v2f*)(&g_lds[c * 4 + 2 * khalf]);       // LDS broadcast
            acc = __builtin_amdgcn_wmma_f32_16x16x4_f32(
                      /*neg_a=*/false, a, /*neg_b=*/false, bb,
                      /*c_mod=*/(short)0, acc,
                      /*reuse_a=*/false, /*reuse_b=*/false);
        }
        // C/D layout: VGPR r -> M=r (lanes 0-15), M=8+r (lanes 16-31); all N
        // columns identical, so lanes with lrow==0 write their 8 rows.
        if (lrow == 0) {
            float* o = out + (size_t)bh * Dd + d0 + 8 * khalf;
            #pragma unroll
            for (int r = 0; r < 8; ++r)
                o[r] = acc[r] * nw_lds[d0 + 8 * khalf + r];
        }
    }
}

extern "C" void kernel_launch(void* const* d_in, const int* in_sizes, int n_in,
                              void* d_out, int out_size, void* d_ws, size_t ws_size,
                              hipStream_t stream) {
    const float* x       = (const float*)d_in[0];  // (4,256,64,2048)
    const float* norm_w  = (const float*)d_in[1];  // (256,)
    const float* score_w = (const float*)d_in[2];  // (256,)
    float* out           = (float*)d_out;          // (4,64,256)

    dim3 grid(4 * Hh);   // one workgroup per (b,h) slice
    dim3 block(256);     // 8 wave32 on one WGP
    axis_attn_pool_kernel<<<grid, block, 0, stream>>>(x, norm_w, score_w, out);
}